// MyLinear_49718541419169
// MI455X (gfx1250) — compile-verified
//
#include <hip/hip_runtime.h>

typedef __attribute__((ext_vector_type(2))) float v2f;
typedef __attribute__((ext_vector_type(8))) float v8f;
typedef int b128_t __attribute__((vector_size(16)));   // matches async builtin param type

#define BM 128
#define BN 128
#define KC 32
#define STR 36   // KC + 4 floats pad: rows stay 16B aligned, ds reads conflict-free

#if __has_builtin(__builtin_amdgcn_global_load_async_to_lds_b128)
#define HAVE_ASYNC 1
#else
#define HAVE_ASYNC 0
#endif

__device__ __forceinline__ void copy_b128(const float* g, float* l) {
#if HAVE_ASYNC
    // async DMA: global -> LDS, tracked by ASYNCcnt, no VGPR round-trip
    __builtin_amdgcn_global_load_async_to_lds_b128(
        (b128_t __attribute__((address_space(1)))*)(g),
        (b128_t __attribute__((address_space(3)))*)(l),
        0, 0);
#else
    *(float4*)l = *(const float4*)g;
#endif
}

#if HAVE_ASYNC
#if __has_builtin(__builtin_amdgcn_s_wait_asynccnt)
#define WAIT_ASYNC(n) __builtin_amdgcn_s_wait_asynccnt(n)
#else
#define WAIT_ASYNC(n) asm volatile("s_wait_asynccnt %0" :: "i"(n) : "memory")
#endif
#else
#define WAIT_ASYNC(n) ((void)0)
#endif

__global__ __launch_bounds__(256)
void linear_wmma_f32(const float* __restrict__ X,    // [M, K]
                     const float* __restrict__ W,    // [N, K]
                     const float* __restrict__ bias, // [N]
                     float* __restrict__ Y,          // [M, N]
                     int M, int N, int K)
{
    __shared__ float Xs[2][BM * STR];
    __shared__ float Ws[2][BN * STR];

    const int tid  = threadIdx.x;
    const int lane = tid & 31;
    const int wave = tid >> 5;

    const int tileM = blockIdx.y * BM;
    const int tileN = blockIdx.x * BN;

    // 8 waves: 2 along M x 4 along N; each wave computes a 64x32 output tile
    const int wm = (wave & 1) * 64;
    const int wn = (wave >> 1) * 32;

    const int ln = lane & 15;   // fragment row / C column
    const int lh = lane >> 4;   // lane half: K pair select / C row half

    v8f acc[4][2] = {};

    // Loop-invariant staging map (identical for X and W tiles since BM == BN):
    // KC=32 -> 8 quads/row, 128 rows -> 1024 quads, 4 per thread.
    int goff[4], loff[4];
    #pragma unroll
    for (int i = 0; i < 4; ++i) {
        int q = tid + i * 256;          // 0..1023
        int r = q >> 3;                 // row 0..127
        int c = (q & 7) << 2;           // col 0,4,...,28
        goff[i] = r * K + c;            // 32-bit per-lane global element offset
        loff[i] = r * STR + c;          // LDS element offset within a buffer
    }
    const float* Xb = X + (size_t)tileM * K;
    const float* Wb = W + (size_t)tileN * K;

    auto issue = [&](int k0, int buf) {
        #pragma unroll
        for (int i = 0; i < 4; ++i)
            copy_b128(Xb + k0 + goff[i], &Xs[buf][loff[i]]);
        #pragma unroll
        for (int i = 0; i < 4; ++i)
            copy_b128(Wb + k0 + goff[i], &Ws[buf][loff[i]]);
    };

    auto compute = [&](int buf) {
        const float* xs = &Xs[buf][0];
        const float* ws = &Ws[buf][0];
        #pragma unroll
        for (int kk = 0; kk < KC; kk += 4) {
            const int kcol = kk + 2 * lh;   // VGPR0: K=kcol, VGPR1: K=kcol+1
            v2f a[4], b[2];
            #pragma unroll
            for (int i = 0; i < 4; ++i)
                a[i] = *(const v2f*)(xs + (wm + 16 * i + ln) * STR + kcol);
            #pragma unroll
            for (int j = 0; j < 2; ++j)
                b[j] = *(const v2f*)(ws + (wn + 16 * j + ln) * STR + kcol);
            #pragma unroll
            for (int i = 0; i < 4; ++i)
                #pragma unroll
                for (int j = 0; j < 2; ++j)
                    acc[i][j] = __builtin_amdgcn_wmma_f32_16x16x4_f32(
                        false, a[i], false, b[j], (short)0, acc[i][j], false, false);
        }
    };

    // Double-buffered pipeline, unrolled x2 so `buf` is a constant everywhere.
    // nchunk = K/KC = 128 (even).
    const int nchunk = K / KC;
    issue(0, 0);
    int ci = 0;
    for (; ci + 2 < nchunk; ci += 2) {
        issue((ci + 1) * KC, 1);
        WAIT_ASYNC(8);                  // chunk ci landed; chunk ci+1 in flight
        __syncthreads();
        compute(0);
        __syncthreads();

        issue((ci + 2) * KC, 0);
        WAIT_ASYNC(8);
        __syncthreads();
        compute(1);
        __syncthreads();
    }
    // tail: chunks ci (buf0, already issued) and ci+1 (buf1)
    issue((ci + 1) * KC, 1);
    WAIT_ASYNC(8);
    __syncthreads();
    compute(0);
    __syncthreads();
    WAIT_ASYNC(0);
    __syncthreads();
    compute(1);

    // Epilogue: C/D layout -> VGPR v : row = v + 8*lh, col = ln.
    // Non-temporal stores: Y is a pure 256MB stream; keep W resident in L2.
    #pragma unroll
    for (int j = 0; j < 2; ++j) {
        const int n    = tileN + wn + j * 16 + ln;
        const float bv = bias[n];
        #pragma unroll
        for (int i = 0; i < 4; ++i) {
            const int mbase = tileM + wm + i * 16 + lh * 8;
            #pragma unroll
            for (int v = 0; v < 8; ++v) {
                __builtin_nontemporal_store(acc[i][j][v] + bv,
                                            &Y[(size_t)(mbase + v) * N + n]);
            }
        }
    }
}

extern "C" void kernel_launch(void* const* d_in, const int* in_sizes, int n_in,
                              void* d_out, int out_size, void* d_ws, size_t ws_size,
                              hipStream_t stream) {
    const float* X  = (const float*)d_in[0];   // x: [8,2048,4096] -> [16384,4096]
    const float* W  = (const float*)d_in[1];   // A: [4096,4096]
    const float* bv = (const float*)d_in[2];   // b: [4096]
    float* Y = (float*)d_out;

    const int N = in_sizes[2];            // 4096
    const int K = in_sizes[1] / N;        // 4096
    const int M = in_sizes[0] / K;        // 16384

    dim3 grid(N / BN, M / BM);            // 32 x 128 blocks
    linear_wmma_f32<<<grid, dim3(256), 0, stream>>>(X, W, bv, Y, M, N, K);
}